// NEC_22926535426059
// MI455X (gfx1250) — compile-verified
//
#include <hip/hip_runtime.h>
#include <hip/hip_bf16.h>

// ---------------------------------------------------------------------------
// NEC forward on MI455X (gfx1250, wave32).
//   1) conv1/conv2/conv3 + relu (direct, fp32)
//   2) fc -> h[32,128]
//   3) distance kernel: for each (action a, 16-key tile), dense fp32 GEMM
//      h[32,128] x keys_a[16,128]^T via V_WMMA_F32_16X16X4_F32, write
//      d2[b,n] = ||h_b||^2 - 2 h_b.k_n + ||k_n||^2 only where actions[b]==a.
//      Tables stream once (82 MB total, L2-resident); no per-sample gather.
//   4) top-50 per sample + inverse-distance weighted sum -> q[32]
// ---------------------------------------------------------------------------

typedef float v2f __attribute__((ext_vector_type(2)));
typedef float v8f __attribute__((ext_vector_type(8)));

#define NEC_B     32
#define NEC_A     8
#define NEC_N     20000
#define NEC_D     128
#define NEC_P     50
#define NEC_DELTA 1e-3f

// -------------------------- conv + relu (direct) ---------------------------
template<int IC, int IH, int IW, int OC, int OH, int OW, int K, int S>
__global__ __launch_bounds__(256)
void nec_conv_relu(const float* __restrict__ in, const float* __restrict__ w,
                   const float* __restrict__ bias, float* __restrict__ out, int total)
{
    int t = blockIdx.x * blockDim.x + threadIdx.x;
    if (t >= total) return;
    int ow = t % OW;
    int oh = (t / OW) % OH;
    int oc = (t / (OW * OH)) % OC;
    int n  = t / (OW * OH * OC);
    const float* ip = in + (size_t)n * IC * IH * IW;
    const float* wp = w + (size_t)oc * IC * K * K;
    float s = bias[oc];
    for (int c = 0; c < IC; ++c)
        for (int kh = 0; kh < K; ++kh)
            #pragma unroll
            for (int kw = 0; kw < K; ++kw)
                s += ip[((size_t)c * IH + oh * S + kh) * IW + ow * S + kw]
                   * wp[(c * K + kh) * K + kw];
    out[t] = s > 0.f ? s : 0.f;
}

// ------------------------------- fc layer ----------------------------------
__global__ __launch_bounds__(256)
void nec_fc(const float* __restrict__ x,   // [32,3136]
            const float* __restrict__ w,   // [128,3136]
            const float* __restrict__ bias,
            float* __restrict__ h)         // [32,128]
{
    int t = blockIdx.x * blockDim.x + threadIdx.x;   // 0..4095
    int d = t & 127;
    int b = t >> 7;
    const float* xp = x + (size_t)b * 3136;
    const float* wp = w + (size_t)d * 3136;
    float s = bias[d];
    for (int i = 0; i < 3136; ++i) s += xp[i] * wp[i];
    h[t] = s;
}

// --------------------- distance GEMM via fp32 WMMA -------------------------
// grid = (N/16, A), block = 32 (one wave). Each block: 16-key tile of action
// a's table vs all 32 batch rows. Two 16x16 accumulators (M-tiles), K-loop of
// 32 steps of V_WMMA_F32_16X16X4_F32 each.
__global__ __launch_bounds__(32)
void nec_dist(const float* __restrict__ h,        // [32,128]
              const int*   __restrict__ actions,  // [32]
              const float* __restrict__ dnd_keys, // [8,20000,128]
              float*       __restrict__ d2)       // [32,20000]
{
    __shared__ float ldsK[16 * NEC_D];   // key tile
    __shared__ float ldsH[32 * NEC_D];   // all batch embeddings
    __shared__ float ldsKK[16];
    __shared__ float ldsHH[32];
    __shared__ int   ldsAct[32];

    const int lane = threadIdx.x;        // 0..31
    const int a    = blockIdx.y;         // action
    const int n0   = blockIdx.x * 16;    // key-tile base

    const int act_l = actions[lane];
    ldsAct[lane] = act_l;
    // Wave-uniform early exit (keeps EXEC all-1s for WMMA afterwards).
    if (__ballot(act_l == a) == 0ull) return;

    // Stage key tile (rows n0..n0+15 are contiguous) and h into LDS, float4.
    const float4* gK = (const float4*)(dnd_keys + ((size_t)a * NEC_N + n0) * NEC_D);
    float4* sK = (float4*)ldsK;
    for (int i = lane; i < 16 * NEC_D / 4; i += 32) sK[i] = gK[i];
    const float4* gH = (const float4*)h;
    float4* sH = (float4*)ldsH;
    for (int i = lane; i < 32 * NEC_D / 4; i += 32) sH[i] = gH[i];
    __syncthreads();

    const int m  = lane & 15;            // A-matrix row / B-matrix column
    const int hi = lane >> 4;            // 0: K pair {0,1}, 1: K pair {2,3}

    v8f acc0 = {};                       // batch rows 0..15
    v8f acc1 = {};                       // batch rows 16..31
    for (int k0 = 0; k0 < NEC_D; k0 += 4) {
        const int k = k0 + 2 * hi;
        // A 16x4 f32 layout: lane m holds row M=m; VGPR0/1 = K = k, k+1.
        v2f a0 = { ldsH[m * NEC_D + k],        ldsH[m * NEC_D + k + 1] };
        v2f a1 = { ldsH[(16 + m) * NEC_D + k], ldsH[(16 + m) * NEC_D + k + 1] };
        // B 4x16 f32 layout: lanes 0-15 rows K=k0,k0+1; lanes 16-31 K=k0+2,3.
        v2f bv = { ldsK[m * NEC_D + k],        ldsK[m * NEC_D + k + 1] };
        acc0 = __builtin_amdgcn_wmma_f32_16x16x4_f32(
                   false, a0, false, bv, (short)0, acc0, false, false);
        acc1 = __builtin_amdgcn_wmma_f32_16x16x4_f32(
                   false, a1, false, bv, (short)0, acc1, false, false);
    }

    // Squared norms from LDS.
    if (lane < 16) {
        float s = 0.f;
        for (int d = 0; d < NEC_D; ++d) { float t = ldsK[lane * NEC_D + d]; s += t * t; }
        ldsKK[lane] = s;
    }
    {
        float s = 0.f;
        for (int d = 0; d < NEC_D; ++d) { float t = ldsH[lane * NEC_D + d]; s += t * t; }
        ldsHH[lane] = s;
    }
    __syncthreads();

    // C/D 16x16 layout: element v of v8f -> M = v + 8*hi, N = lane&15.
    const int n = n0 + m;
    #pragma unroll
    for (int v = 0; v < 8; ++v) {
        int b0 = v + hi * 8;
        if (ldsAct[b0] == a)
            d2[(size_t)b0 * NEC_N + n] = ldsHH[b0] - 2.f * acc0[v] + ldsKK[m];
        int b1 = 16 + v + hi * 8;
        if (ldsAct[b1] == a)
            d2[(size_t)b1 * NEC_N + n] = ldsHH[b1] - 2.f * acc1[v] + ldsKK[m];
    }
}

// ------------------- top-P + inverse-distance weighting --------------------
// One block per sample. 50 deterministic argmin passes (smallest-index
// tie-break), excluding selected entries by writing +INF into d2 (d2 is
// fully regenerated by nec_dist every call, so this is replay-safe).
__global__ __launch_bounds__(256)
void nec_topk(float* __restrict__ d2,              // [32,20000] (scratch, mutated)
              const int*   __restrict__ actions,   // [32]
              const float* __restrict__ dnd_values,// [8,20000]
              float* __restrict__ q)               // [32]
{
    __shared__ float rv[256];
    __shared__ int   ri[256];
    __shared__ float selV[NEC_P];
    __shared__ int   selI[NEC_P];

    const int b = blockIdx.x;
    const int t = threadIdx.x;
    float* row = d2 + (size_t)b * NEC_N;

    for (int p = 0; p < NEC_P; ++p) {
        float best = INFINITY; int bidx = NEC_N;
        for (int n = t; n < NEC_N; n += 256) {
            float v = row[n];
            if (v < best || (v == best && n < bidx)) { best = v; bidx = n; }
        }
        rv[t] = best; ri[t] = bidx;
        __syncthreads();
        if (t == 0) {
            float gb = rv[0]; int gi = ri[0];
            for (int i = 1; i < 256; ++i)
                if (rv[i] < gb || (rv[i] == gb && ri[i] < gi)) { gb = rv[i]; gi = ri[i]; }
            selV[p] = gb; selI[p] = gi;
            row[gi] = INFINITY;          // exclude from later passes
        }
        __syncthreads();                 // makes the global INF visible in-block
    }

    if (t == 0) {
        const float* vals = dnd_values + (size_t)actions[b] * NEC_N;
        float num = 0.f, den = 0.f;
        for (int p = 0; p < NEC_P; ++p) {
            float dd = selV[p]; if (dd < 0.f) dd = 0.f;
            float w = 1.0f / (dd + NEC_DELTA);
            num += w * vals[selI[p]];
            den += w;
        }
        q[b] = num / den;
    }
}

// ------------------------------- launcher ----------------------------------
extern "C" void kernel_launch(void* const* d_in, const int* in_sizes, int n_in,
                              void* d_out, int out_size, void* d_ws, size_t ws_size,
                              hipStream_t stream)
{
    (void)in_sizes; (void)n_in; (void)out_size; (void)ws_size;

    const float* states = (const float*)d_in[0];
    const int*   actions= (const int*)  d_in[1];
    const float* c1w = (const float*)d_in[2];
    const float* c1b = (const float*)d_in[3];
    const float* c2w = (const float*)d_in[4];
    const float* c2b = (const float*)d_in[5];
    const float* c3w = (const float*)d_in[6];
    const float* c3b = (const float*)d_in[7];
    const float* fcw = (const float*)d_in[8];
    const float* fcb = (const float*)d_in[9];
    const float* keys= (const float*)d_in[10];
    const float* vals= (const float*)d_in[11];
    float* q = (float*)d_out;

    // Workspace layout (floats): conv1 | conv2 | conv3 | h | d2  (~5.3 MB)
    float* ws = (float*)d_ws;
    float* x1 = ws;                         // 32*32*20*20 = 409600
    float* x2 = x1 + 409600;                // 32*64*9*9   = 165888
    float* x3 = x2 + 165888;                // 32*64*7*7   = 100352
    float* h  = x3 + 100352;                // 32*128      = 4096
    float* d2 = h  + 4096;                  // 32*20000    = 640000

    nec_conv_relu<4, 84, 84, 32, 20, 20, 8, 4>
        <<<(409600 + 255) / 256, 256, 0, stream>>>(states, c1w, c1b, x1, 409600);
    nec_conv_relu<32, 20, 20, 64, 9, 9, 4, 2>
        <<<(165888 + 255) / 256, 256, 0, stream>>>(x1, c2w, c2b, x2, 165888);
    nec_conv_relu<64, 9, 9, 64, 7, 7, 3, 1>
        <<<(100352 + 255) / 256, 256, 0, stream>>>(x2, c3w, c3b, x3, 100352);

    nec_fc<<<16, 256, 0, stream>>>(x3, fcw, fcb, h);

    nec_dist<<<dim3(NEC_N / 16, NEC_A), 32, 0, stream>>>(h, actions, keys, d2);

    nec_topk<<<NEC_B, 256, 0, stream>>>(d2, actions, vals, q);
}